// SoftMD5_49933289783693
// MI455X (gfx1250) — compile-verified
//
#include <hip/hip_runtime.h>
#include <stdint.h>

// ---------------------------------------------------------------------------
// SoftMD5 forward == exact MD5 with message words W_g = floor(sum w_j * 2^j)
// (float64 accumulation, matching the reference's _hard_add semantics).
// Data path: TDM tensor_load_to_lds stages 32-row x 128-byte tiles per wave,
// double-buffered, with the TDM pad feature producing a 33-dword LDS row
// stride (bank-conflict-free for 32 lanes). Per-wave s_wait_tensorcnt sync.
// ---------------------------------------------------------------------------

typedef unsigned int u32x4 __attribute__((ext_vector_type(4)));
typedef int          i32x8 __attribute__((ext_vector_type(8)));
typedef int          i32x4 __attribute__((ext_vector_type(4)));

#if defined(__has_builtin)
#if __has_builtin(__builtin_amdgcn_tensor_load_to_lds)
#define HAVE_TDM 1
#endif
#endif
#ifndef HAVE_TDM
#define HAVE_TDM 0
#endif

static constexpr uint32_t MD5K[64] = {
    0xd76aa478u, 0xe8c7b756u, 0x242070dbu, 0xc1bdceeeu,
    0xf57c0fafu, 0x4787c62au, 0xa8304613u, 0xfd469501u,
    0x698098d8u, 0x8b44f7afu, 0xffff5bb1u, 0x895cd7beu,
    0x6b901122u, 0xfd987193u, 0xa679438eu, 0x49b40821u,
    0xf61e2562u, 0xc040b340u, 0x265e5a51u, 0xe9b6c7aau,
    0xd62f105du, 0x02441453u, 0xd8a1e681u, 0xe7d3fbc8u,
    0x21e1cde6u, 0xc33707d6u, 0xf4d50d87u, 0x455a14edu,
    0xa9e3e905u, 0xfcefa3f8u, 0x676f02d9u, 0x8d2a4c8au,
    0xfffa3942u, 0x8771f681u, 0x6d9d6122u, 0xfde5380cu,
    0xa4beea44u, 0x4bdecfa9u, 0xf6bb4b60u, 0xbebfbc70u,
    0x289b7ec6u, 0xeaa127fau, 0xd4ef3085u, 0x04881d05u,
    0xd9d4d039u, 0xe6db99e5u, 0x1fa27cf8u, 0xc4ac5665u,
    0xf4292244u, 0x432aff97u, 0xab9423a7u, 0xfc93a039u,
    0x655b59c3u, 0x8f0ccc92u, 0xffeff47du, 0x85845dd1u,
    0x6fa87e4fu, 0xfe2ce6e0u, 0xa3014314u, 0x4e0811a1u,
    0xf7537e82u, 0xbd3af235u, 0x2ad7d2bbu, 0xeb86d391u};

static constexpr int MD5S[64] = {
    7, 12, 17, 22, 7, 12, 17, 22, 7, 12, 17, 22, 7, 12, 17, 22,
    5, 9, 14, 20, 5, 9, 14, 20, 5, 9, 14, 20, 5, 9, 14, 20,
    4, 11, 16, 23, 4, 11, 16, 23, 4, 11, 16, 23, 4, 11, 16, 23,
    6, 10, 15, 21, 6, 10, 15, 21, 6, 10, 15, 21, 6, 10, 15, 21};

__device__ __forceinline__ uint32_t rotl32(uint32_t x, int s) {
    return (x << s) | (x >> (32 - s));
}

__device__ __forceinline__ void md5_compress(uint32_t st[4], const uint32_t w[16]) {
    uint32_t a = st[0], b = st[1], c = st[2], d = st[3];
#pragma unroll
    for (int i = 0; i < 64; ++i) {
        uint32_t f;
        int g;
        if (i < 16) {
            f = (b & c) | (~b & d);
            g = i;
        } else if (i < 32) {
            f = (d & b) | (~d & c);
            g = (5 * i + 1) & 15;
        } else if (i < 48) {
            f = b ^ c ^ d;
            g = (3 * i + 5) & 15;
        } else {
            f = c ^ (b | ~d);
            g = (7 * i) & 15;
        }
        f = f + a + MD5K[i] + w[g];
        a = d;
        d = c;
        c = b;
        b = b + rotl32(f, MD5S[i]);
    }
    st[0] += a;
    st[1] += b;
    st[2] += c;
    st[3] += d;
}

// Horner in f64: acc = sum_{j=0..31} p[j] * 2^j ; floor via u64 truncation.
__device__ __forceinline__ uint32_t word_from_bits(const float* p) {
    double acc = 0.0;
#pragma unroll
    for (int j = 31; j >= 0; --j)
        acc = __builtin_fma(acc, 2.0, (double)p[j]);
    return (uint32_t)(unsigned long long)acc;
}

#if HAVE_TDM
// One 2D TDM tile: 32 rows x 32 f32 (128B), row stride 512 f32 in memory,
// LDS pad: +1 dword every 32 dwords -> 33-dword LDS row stride.
__device__ __forceinline__ void tdm_issue(uint32_t lds_byte, uint64_t gaddr) {
    uint32_t lds_s = (uint32_t)__builtin_amdgcn_readfirstlane((int)lds_byte);
    uint32_t glo = (uint32_t)__builtin_amdgcn_readfirstlane((int)(uint32_t)gaddr);
    uint32_t ghi = (uint32_t)__builtin_amdgcn_readfirstlane((int)(uint32_t)(gaddr >> 32));
    // D# group 0: count=1 | lds_addr | global_addr(57b) | type=2
    u32x4 g0 = {1u, lds_s, glo, (ghi & 0x01FFFFFFu) | 0x80000000u};
    // D# group 1:
    //  w0: data_size=2(4B)<<16 | pad_enable<<20 | pad_interval=4(32dw)<<22 | pad_amount=0(1dw)<<25
    //  w1: tensor_dim0(=512) low16 -> [31:16]
    //  w2: tensor_dim0 hi16 | tensor_dim1 low16      (dim1 = 2^30)
    //  w3: tensor_dim1 hi16(0x4000) | tile_dim0(32)<<16
    //  w4: tile_dim1(32) | tile_dim2(0)<<16
    //  w5: tensor_dim0_stride low32 = 512
    //  w6,w7: stride hi / dim1 stride (unused, 0)
    i32x8 g1 = {(int)0x01120000u,
                (int)(512u << 16),
                0,
                (int)(0x4000u | (32u << 16)),
                32,
                512,
                0,
                0};
    i32x4 gz4 = {0, 0, 0, 0};
    i32x8 gz8 = {0, 0, 0, 0, 0, 0, 0, 0};
    // clang-23 / amdgpu-toolchain arity: (v4u, v8i, v4i, v4i, v8i, i32 cpol)
    __builtin_amdgcn_tensor_load_to_lds(g0, g1, gz4, gz4, gz8, 0);
}
#endif

#define ROWS_PER_BLOCK 128
#define LDS_ROW_DW 33            // 32 data dwords + 1 pad dword
#define WAVE_TILE_F (32 * LDS_ROW_DW)  // 1056 floats per wave per parity
#define WAVE_TILE_B (WAVE_TILE_F * 4)  // 4224 bytes
#define PARITY_F (4 * WAVE_TILE_F)     // 4224 floats
#define PARITY_B (4 * WAVE_TILE_B)     // 16896 bytes

__global__ __launch_bounds__(ROWS_PER_BLOCK) void softmd5_kernel(
    const float* __restrict__ msg, float* __restrict__ out, int B) {
    __shared__ float smem[2 * PARITY_F];  // 33792 bytes

    const int t = threadIdx.x;
    const int wave = t >> 5;
    const int lane = t & 31;
    const long long blockRow0 = (long long)blockIdx.x * ROWS_PER_BLOCK;
    const long long row = blockRow0 + t;
    const long long waveRow0 = blockRow0 + (long long)wave * 32;
    const bool valid = row < (long long)B;

    uint32_t wv[16];

#if HAVE_TDM
    if (waveRow0 + 32 <= (long long)B) {
        const uint64_t base =
            (uint64_t)(uintptr_t)msg + (uint64_t)waveRow0 * 2048ull;  // 512 f32/row
        const uint32_t lds0 =
            (uint32_t)(uintptr_t)smem + (uint32_t)wave * (uint32_t)WAVE_TILE_B;

        tdm_issue(lds0, base);  // chunk 0 -> parity 0
#pragma unroll
        for (int g = 0; g < 16; ++g) {
            if (g + 1 < 16) {
                // ensure previous reads of the target buffer are done before
                // the TDM overwrites it (TDM writes are unordered vs DS reads)
                asm volatile("s_wait_dscnt 0" ::: "memory");
                tdm_issue(lds0 + (uint32_t)(((g + 1) & 1) * PARITY_B),
                          base + (uint64_t)(g + 1) * 128ull);
                __builtin_amdgcn_s_wait_tensorcnt(1);  // chunk g done, prefetch in flight
            } else {
                __builtin_amdgcn_s_wait_tensorcnt(0);
            }
            const float* p =
                &smem[((g & 1) ? PARITY_F : 0) + wave * WAVE_TILE_F + lane * LDS_ROW_DW];
            wv[g] = word_from_bits(p);
        }
    } else
#endif
    {
        if (valid) {
            const float* rp = msg + (long long)row * 512;
#pragma unroll
            for (int g = 0; g < 16; ++g) {
                float tmp[32];
#pragma unroll
                for (int j = 0; j < 32; ++j) tmp[j] = rp[g * 32 + j];
                wv[g] = word_from_bits(tmp);
            }
        }
    }

    if (!valid) return;

    uint32_t st[4] = {0x67452301u, 0xefcdab89u, 0x98badcfeu, 0x10325476u};
    md5_compress(st, wv);

    // Second block: standard MD5 padding for a 512-bit message (constants).
    const uint32_t w2[16] = {0x80u, 0, 0, 0, 0, 0, 0, 0,
                             0,     0, 0, 0, 0, 0, 0x200u, 0};
    md5_compress(st, w2);

    // Emit 128 bits (a,b,c,d LSB-first) as floats, float4-vectorized.
    float4* o = (float4*)(out + (size_t)row * 128);
#pragma unroll
    for (int k = 0; k < 4; ++k) {
        const uint32_t v = st[k];
#pragma unroll
        for (int j = 0; j < 32; j += 4) {
            float4 f4;
            f4.x = (float)((v >> (j + 0)) & 1u);
            f4.y = (float)((v >> (j + 1)) & 1u);
            f4.z = (float)((v >> (j + 2)) & 1u);
            f4.w = (float)((v >> (j + 3)) & 1u);
            o[k * 8 + (j >> 2)] = f4;
        }
    }
}

extern "C" void kernel_launch(void* const* d_in, const int* in_sizes, int n_in,
                              void* d_out, int out_size, void* d_ws, size_t ws_size,
                              hipStream_t stream) {
    (void)n_in;
    (void)out_size;
    (void)d_ws;
    (void)ws_size;
    const float* msg = (const float*)d_in[0];
    float* out = (float*)d_out;
    const int B = in_sizes[0] / 512;
    if (B <= 0) return;
    const int grid = (B + ROWS_PER_BLOCK - 1) / ROWS_PER_BLOCK;
    softmd5_kernel<<<grid, ROWS_PER_BLOCK, 0, stream>>>(msg, out, B);
}